// diff_stft_24412594110628
// MI455X (gfx1250) — compile-verified
//
#include <hip/hip_runtime.h>

// ---------------------------------------------------------------------------
// reference(out, y) = mean((out-y)^2) + sum( (Re STFT(out) - Re STFT(y))^2 )
// STFT is linear  =>  work on d = out - y only.
//   WIN=512, HOP=256, periodic Hann, rfft / sum(win)=256, 257 real bins,
//   16 batches x 1,048,576 samples, zero pad 256 both ends -> 4097 frames.
// Bins 0..255: GEMM [frames x 512] x [512 x 256 windowed-cos basis] in fp32
//   -> V_WMMA_F32_16X16X4_F32, 16 bin tiles = 4 per wave (balanced, no branch).
// Bin 256 (Nyquist): basis is (-1)^n -> trivial per-frame VALU dot product.
// Memory floor ~128 MiB @ 23.3 TB/s; table (512 KiB) stays L2-resident.
// ---------------------------------------------------------------------------

typedef __attribute__((ext_vector_type(2))) float v2f;
typedef __attribute__((ext_vector_type(8))) float v8f;

#define WIN        512
#define HOP        256
#define NCOL       256              // GEMM columns = bins 0..255 (16 tiles)
#define TLEN       1048576
#define NBATCH     16
#define FT_PER_B   257              // ceil(4097 frames / 16)
#define NPART      (NBATCH * FT_PER_B)   // 4112 workgroup partials
#define PITCH      260              // LDS row pitch (floats): 260%64==4 -> conflict-free
#define TAB_ELEMS  (WIN * NCOL)     // 131072 floats = 512 KiB
#define W0F        (6.283185307179586f / 512.0f)   // 2*pi/512

// ---------------------------------------------------------------------------
// Window-folded cosine table, pair-interleaved over n so a lane's two K-rows
// (n=2p, 2p+1) for a given column are one aligned 8-byte global load.
//   tab2[ ((p*256 + c) << 1) + (n&1) ] = w[n] * cos(2*pi*n*c/512),  c in 0..255
// ---------------------------------------------------------------------------
__global__ void build_tab_kernel(float* __restrict__ tab2) {
  int idx = blockIdx.x * blockDim.x + threadIdx.x;
  if (idx >= TAB_ELEMS) return;
  int n = idx / NCOL;
  int c = idx % NCOL;
  float wn = 0.5f - 0.5f * __cosf(W0F * (float)n);  // periodic Hann
  int p = (n * c) & 511;                            // keep cos argument small
  float v = wn * __cosf(W0F * (float)p);
  tab2[(((n >> 1) * NCOL + c) << 1) + (n & 1)] = v;
}

// ---------------------------------------------------------------------------
// One workgroup = (batch b, tile t of 16 consecutive frames).
// 128 threads = 4 waves; wave w owns bin tiles {w, w+4, w+8, w+12}.
// ---------------------------------------------------------------------------
__global__ void __launch_bounds__(128) stft_mse_kernel(
    const float* __restrict__ xout, const float* __restrict__ xy,
    const float* __restrict__ tab2, float* __restrict__ partials) {
  __shared__ __align__(16) float dlds[17 * PITCH];  // 17 rows x 256 samples (pitch 260)
  __shared__ __align__(16) float nyqw[WIN];         // w[n] * (-1)^n  (Nyquist basis)
  __shared__ float red[128];

  const int t   = blockIdx.x;          // frame tile (0..256)
  const int b   = blockIdx.y;          // batch
  const int tid = threadIdx.x;

  const float* ob = xout + (size_t)b * TLEN;
  const float* yb = xy   + (size_t)b * TLEN;

  // Padded-signal index of local sample i is t*4096 + i; unpadded j = that-256.
  const long base = (long)t * 4096 - 256;

  // Stage d = out - y (zero outside data: covers zero-padding AND makes
  // out-of-range frames of the last tile contribute exactly 0).
  // Fuse diff1: each i<4096 maps to a unique global sample across tiles.
  float mse = 0.0f;
  for (int i = tid; i < 17 * 256; i += 128) {   // 34 iters, no divergence
    long j = base + (long)i;
    float d = 0.0f;
    if ((unsigned long)j < (unsigned long)TLEN) d = ob[j] - yb[j];
    dlds[(i >> 8) * PITCH + (i & 255)] = d;
    if (i < 4096) mse += d * d;
  }
  for (int n = tid; n < WIN; n += 128) {        // Nyquist basis: w[n]*(-1)^n
    float wn = 0.5f - 0.5f * __cosf(W0F * (float)n);
    nyqw[n] = (n & 1) ? -wn : wn;
  }
  __syncthreads();

  const int lane = tid & 31;
  const int wave = tid >> 5;
  const int m    = lane & 15;          // A: frame row / B: column lane
  const int kb   = (lane >> 4) * 2;    // A: K sub-index / B: K row base

  v8f acc[4] = {};                     // 4 live 16x16 f32 accumulators per wave

  // K-loop split at the LDS row boundary (K=256): within a half both the
  // A address (ap + K2) and the table address (tbase + K2*256) are linear
  // in K2 -> strength-reduced bases + immediate offsets, no per-step VALU.
#pragma unroll
  for (int half = 0; half < 2; ++half) {
    // A fragment 16x4 f32 (ISA layout): lanes0-15 K=0..1, lanes16-31 K=2..3.
    // Even column + pitch => aligned ds_load_b64, conflict-free banks.
    const float* ap    = &dlds[(m + half) * PITCH + kb];
    const float* tbase = tab2 + ((size_t)(half * 128 + (kb >> 1)) << 9);  // *512
    for (int K2 = 0; K2 < 256; K2 += 4) {
      v2f a = *(const v2f*)&ap[K2];
      const float* trow = tbase + ((size_t)K2 << 8);  // pair row, K2/2 * 512
#pragma unroll
      for (int jt = 0; jt < 4; ++jt) { // bin tiles wave, wave+4, wave+8, wave+12
        int c = (wave + 4 * jt) * 16 + m;
        v2f bv = *(const v2f*)&trow[c << 1];   // B rows (K+kb, K+kb+1), col c
        acc[jt] = __builtin_amdgcn_wmma_f32_16x16x4_f32(
            /*neg_a=*/false, a, /*neg_b=*/false, bv,
            /*c_mod=*/(short)0, acc[jt], /*reuse_a=*/false, /*reuse_b=*/false);
      }
    }
  }

  // Sum of squares of D is layout-agnostic: every tile element appears once.
  float s = 0.0f;
#pragma unroll
  for (int jt = 0; jt < 4; ++jt) {
#pragma unroll
    for (int r = 0; r < 8; ++r) s += acc[jt][r] * acc[jt][r];
  }

  // Nyquist bin: Re(X_256)[frame] = sum_n w[n]*(-1)^n * d[frame*256+n].
  // 8 threads per frame; frame row is fixed, so split at the n=256 row
  // boundary and walk with constant bases + immediate offsets (v2f pairs ->
  // ds_load_2addr_b64, no per-element address math). Shuffle-combine in wave.
  {
    const int mf = tid >> 3;           // frame 0..15
    const int l8 = (tid & 7) * 2;      // even column start -> aligned pairs
    const float* row0 = &dlds[mf * PITCH + l8];
    const float* row1 = &dlds[(mf + 1) * PITCH + l8];
    const float* wv0  = &nyqw[l8];
    float nyq = 0.0f;
#pragma unroll
    for (int k = 0; k < 16; ++k) {     // n = l8 + 16k        (first half)
      v2f dv = *(const v2f*)&row0[16 * k];
      v2f wv = *(const v2f*)&wv0[16 * k];
      nyq += dv.x * wv.x + dv.y * wv.y;
    }
#pragma unroll
    for (int k = 0; k < 16; ++k) {     // n = 256 + l8 + 16k  (second half)
      v2f dv = *(const v2f*)&row1[16 * k];
      v2f wv = *(const v2f*)&wv0[256 + 16 * k];
      nyq += dv.x * wv.x + dv.y * wv.y;
    }
    nyq += __shfl_down(nyq, 4, 32);
    nyq += __shfl_down(nyq, 2, 32);
    nyq += __shfl_down(nyq, 1, 32);
    if ((tid & 7) == 0) s += nyq * nyq;  // same unnormalized scale as D
  }

  // diff2 scale: (D/256)^2 -> /65536 ; diff1 scale: /(16*1048576)
  red[tid] = s * (1.0f / 65536.0f) + mse * (1.0f / 16777216.0f);
  __syncthreads();
  for (int off = 64; off > 0; off >>= 1) {
    if (tid < off) red[tid] += red[tid + off];
    __syncthreads();
  }
  if (tid == 0) partials[b * FT_PER_B + t] = red[0];
}

// Fixed-shape, fixed-order reduction => bitwise deterministic across replays.
__global__ void __launch_bounds__(256) final_reduce_kernel(
    const float* __restrict__ partials, float* __restrict__ dout) {
  __shared__ float red[256];
  int tid = threadIdx.x;
  float s = 0.0f;
  for (int i = tid; i < NPART; i += 256) s += partials[i];
  red[tid] = s;
  __syncthreads();
  for (int off = 128; off > 0; off >>= 1) {
    if (tid < off) red[tid] += red[tid + off];
    __syncthreads();
  }
  if (tid == 0) dout[0] = red[0];
}

extern "C" void kernel_launch(void* const* d_in, const int* in_sizes, int n_in,
                              void* d_out, int out_size, void* d_ws, size_t ws_size,
                              hipStream_t stream) {
  const float* xout = (const float*)d_in[0];
  const float* xy   = (const float*)d_in[1];
  float* tab2     = (float*)d_ws;              // 512*256 floats = 512 KiB
  float* partials = tab2 + TAB_ELEMS;          // 4112 floats

  build_tab_kernel<<<(TAB_ELEMS + 255) / 256, 256, 0, stream>>>(tab2);

  dim3 grid(FT_PER_B, NBATCH);                 // 257 x 16 workgroups
  stft_mse_kernel<<<grid, 128, 0, stream>>>(xout, xy, tab2, partials);

  final_reduce_kernel<<<1, 256, 0, stream>>>(partials, (float*)d_out);
}